// custom_loss_50508815400972
// MI455X (gfx1250) — compile-verified
//
#include <hip/hip_runtime.h>

// ---------------------------------------------------------------------------
// loss = sum_b mean_{h,w in [5,2042]} ( s1*s2 - 2*s12 )
// where each filtered map is a vertical 3-tap conv (taps K0,K1,K0) because the
// reference's 3x3 conv collapses along the size-1 channel dim.
// Memory-bound: 256 MiB streamed once -> ~11.5 us floor @ 23.3 TB/s.
// CDNA5 path: async global->LDS DMA staging (ASYNCcnt) + wave32 blocks.
// ---------------------------------------------------------------------------

namespace {
constexpr int   TILE_W = 256;        // columns per block (1 per thread)
constexpr int   ROWS   = 30;         // output rows per block
constexpr int   INROWS = ROWS + 2;   // staged input rows (halo +-1)
constexpr int   CXN    = 8;          // column tiles  (8*256 >= 2038 cols)
constexpr int   RYN    = 68;         // row tiles     (68*30 >= 2038 rows)
constexpr int   IMGS   = 8;
constexpr int   NBLK   = IMGS * CXN * RYN;   // 4352 blocks
constexpr int   H = 2048, W = 2048;
constexpr float K0 = 0.11831801f;    // kernel[1,0] == kernel[1,2]
constexpr float K1 = 0.14776132f;    // kernel[1,1]
}

__global__ __launch_bounds__(256) void ssim_loss_main(
    const float* __restrict__ X, const float* __restrict__ Y,
    float* __restrict__ partial)
{
  __shared__ float lx[INROWS * TILE_W];   // 32 KB
  __shared__ float ly[INROWS * TILE_W];   // 32 KB
  __shared__ float red[256];

  const int t  = threadIdx.x;
  int bx = blockIdx.x;
  const int cx = bx % CXN; bx /= CXN;
  const int ry = bx % RYN; bx /= RYN;
  const int b  = bx;                         // image index 0..7

  const int w0a = 4 + TILE_W * cx;           // 16B-aligned column base
  const int h0  = 5 + ROWS * ry;             // first output row of this block

  const float* Xb = X + (size_t)b * (size_t)(H * W);
  const float* Yb = Y + (size_t)b * (size_t)(H * W);

  // __shared__ flat address: low 32 bits are the LDS byte offset.
  const unsigned lxb = (unsigned)(size_t)(&lx[0]);
  const unsigned lyb = (unsigned)(size_t)(&ly[0]);

  // ---- Stage 32 rows x 256 cols of X and Y into LDS via async DMA --------
  // 2 arrays * 32 rows * 64 16B-packets = 4096 packets; 16 per thread.
  // arr is block-uniform per k, so the SGPR base is uniform.
#pragma unroll
  for (int k = 0; k < 16; ++k) {
    const int arr = k >> 3;                      // 0 = X, 1 = Y
    const int q   = ((k & 7) << 8) + t;          // packet id within array
    const int i   = q >> 6;                      // staged row 0..31
    const int pc  = q & 63;                      // 16B packet within row
    const unsigned gb = 4u * ((unsigned)(h0 - 1 + i) * (unsigned)W +
                              (unsigned)(w0a + (pc << 2)));
    const unsigned lo = (arr ? lyb : lxb) +
                        4u * (unsigned)((i << 8) + (pc << 2));
    const float* base = arr ? Yb : Xb;
    asm volatile("global_load_async_to_lds_b128 %0, %1, %2"
                 :: "v"(lo), "v"(gb), "s"(base) : "memory");
  }
  asm volatile("s_wait_asynccnt 0" ::: "memory");
  __syncthreads();

  // ---- Compute: sliding 3-row register window per column -----------------
  const int  w    = w0a + t;
  const bool wval = (w >= 5) && (w <= W - 6);

  float x0 = lx[t],          x1 = lx[TILE_W + t];
  float y0 = ly[t],          y1 = ly[TILE_W + t];
  float xx0 = x0 * x0, xx1 = x1 * x1;
  float yy0 = y0 * y0, yy1 = y1 * y1;
  float xy0 = x0 * y0, xy1 = x1 * y1;

  float acc = 0.f;
#pragma unroll
  for (int j = 0; j < ROWS; ++j) {
    const float x2 = lx[(j + 2) * TILE_W + t];
    const float y2 = ly[(j + 2) * TILE_W + t];
    const float xx2 = x2 * x2, yy2 = y2 * y2, xy2 = x2 * y2;

    const float mu1 = fmaf(K1, x1,  K0 * (x0 + x2));
    const float mu2 = fmaf(K1, y1,  K0 * (y0 + y2));
    const float s1  = fmaf(K1, xx1, K0 * (xx0 + xx2)) - mu1 * mu1;
    const float s2  = fmaf(K1, yy1, K0 * (yy0 + yy2)) - mu2 * mu2;
    const float s12 = fmaf(K1, xy1, K0 * (xy0 + xy2)) - mu1 * mu2;
    const float loss = fmaf(s1, s2, -2.f * s12);

    if (wval && (h0 + j) <= (H - 6)) acc += loss;

    x0 = x1; x1 = x2;   y0 = y1; y1 = y2;
    xx0 = xx1; xx1 = xx2;  yy0 = yy1; yy1 = yy2;  xy0 = xy1; xy1 = xy2;
  }

  // ---- Block reduction (deterministic) -----------------------------------
  red[t] = acc;
  __syncthreads();
#pragma unroll
  for (int s = 128; s > 0; s >>= 1) {
    if (t < s) red[t] += red[t + s];
    __syncthreads();
  }
  if (t == 0) partial[blockIdx.x] = red[0];
}

__global__ __launch_bounds__(256) void ssim_loss_reduce(
    const float* __restrict__ partial, float* __restrict__ out, int n)
{
  __shared__ float red[256];
  const int t = threadIdx.x;
  float a = 0.f;
  for (int i = t; i < n; i += 256) a += partial[i];   // fixed order
  red[t] = a;
  __syncthreads();
#pragma unroll
  for (int s = 128; s > 0; s >>= 1) {
    if (t < s) red[t] += red[t + s];
    __syncthreads();
  }
  if (t == 0) out[0] = red[0] * (1.0f / (2038.0f * 2038.0f));
}

extern "C" void kernel_launch(void* const* d_in, const int* in_sizes, int n_in,
                              void* d_out, int out_size, void* d_ws, size_t ws_size,
                              hipStream_t stream) {
  (void)in_sizes; (void)n_in; (void)out_size; (void)ws_size;
  const float* X = (const float*)d_in[0];
  const float* Y = (const float*)d_in[1];
  float* out = (float*)d_out;
  float* ws  = (float*)d_ws;       // needs NBLK*4 = 17,408 bytes

  ssim_loss_main<<<NBLK, 256, 0, stream>>>(X, Y, ws);
  ssim_loss_reduce<<<1, 256, 0, stream>>>(ws, out, NBLK);
}